// VGGBlock_67980742361312
// MI455X (gfx1250) — compile-verified
//
#include <hip/hip_runtime.h>
#include <hip/hip_bf16.h>
#include <math.h>

// ---------------------------------------------------------------------------
// VGGBlock sparse-conv for MI455X (gfx1250, wave32, WMMA + async-LDS).
//   BN(stats) -> BN(apply)+ReLU -> 27x gather-GEMM (N x 32) @ (32 x 64) via
//   V_WMMA_F32_16X16X4_F32, with W slices staged by GLOBAL_LOAD_ASYNC_TO_LDS
//   into a double-buffered LDS tile (DMA overlaps WMMA compute).
// ---------------------------------------------------------------------------

typedef __attribute__((ext_vector_type(2))) float v2f;
typedef __attribute__((ext_vector_type(8))) float v8f;

#define C_IN   32
#define C_OUT  64
#define KOFF   27
#define RB     512          // reduction blocks for BN stats
#define BN_EPS 1e-5f

// ---------------- Kernel 1: per-block BN partial sums ----------------------
__global__ __launch_bounds__(256) void bn_partials(const float* __restrict__ f,
                                                   float* __restrict__ partials,
                                                   int N) {
    __shared__ float ss[256];
    __shared__ float sq[256];
    const int tid = threadIdx.x;
    const size_t total = (size_t)N * C_IN;
    float s = 0.f, q = 0.f;
    // stride is a multiple of 32 -> each lane always touches channel (tid & 31)
    for (size_t i = (size_t)blockIdx.x * 256 + tid; i < total;
         i += (size_t)gridDim.x * 256) {
        float v = f[i];
        s += v;
        q += v * v;
    }
    ss[tid] = s;
    sq[tid] = q;
    __syncthreads();
    if (tid < C_IN) {
        float ts = 0.f, tq = 0.f;
#pragma unroll
        for (int k = 0; k < 8; ++k) {
            ts += ss[tid + 32 * k];
            tq += sq[tid + 32 * k];
        }
        partials[blockIdx.x * 64 + tid] = ts;
        partials[blockIdx.x * 64 + 32 + tid] = tq;
    }
}

// ---------------- Kernel 2: finalize scale/bias ----------------------------
__global__ void bn_finalize(const float* __restrict__ partials,
                            const float* __restrict__ gamma,
                            const float* __restrict__ beta,
                            float* __restrict__ sb, int N) {
    const int c = threadIdx.x;
    if (c < C_IN) {
        float s = 0.f, q = 0.f;
        for (int b = 0; b < RB; ++b) {
            s += partials[b * 64 + c];
            q += partials[b * 64 + 32 + c];
        }
        const float invN  = 1.0f / (float)N;
        const float mean  = s * invN;
        const float var   = q * invN - mean * mean;
        const float inv   = rsqrtf(var + BN_EPS);
        const float scale = gamma[c] * inv;
        sb[c]      = scale;                      // y = scale*x + bias
        sb[32 + c] = beta[c] - mean * scale;
    }
}

// ---------------- Kernel 3: normalize + ReLU into workspace ----------------
__global__ __launch_bounds__(256) void bn_apply(const float* __restrict__ f,
                                                const float* __restrict__ sb,
                                                float* __restrict__ xn, int N) {
    __shared__ float s[64];
    if (threadIdx.x < 64) s[threadIdx.x] = sb[threadIdx.x];
    __syncthreads();
    const size_t total4 = (size_t)N * (C_IN / 4);
    const size_t i = (size_t)blockIdx.x * 256 + threadIdx.x;
    if (i < total4) {
        const float4 v = ((const float4*)f)[i];
        const int c = (int)((i * 4) & (C_IN - 1));
        float4 r;
        r.x = fmaxf(fmaf(v.x, s[c + 0], s[32 + c + 0]), 0.f);
        r.y = fmaxf(fmaf(v.y, s[c + 1], s[32 + c + 1]), 0.f);
        r.z = fmaxf(fmaf(v.z, s[c + 2], s[32 + c + 2]), 0.f);
        r.w = fmaxf(fmaf(v.w, s[c + 3], s[32 + c + 3]), 0.f);
        ((float4*)xn)[i] = r;
    }
}

// Issue async DMA of one 8 KB weight slice (27x32x64 f32 layout) into LDS.
// 256 threads x 2 x 16 B = 8192 B. INST_OFFSET is added to BOTH the global
// and LDS addresses, so one offset:4096 instruction covers the second half.
__device__ __forceinline__ void stage_w_async(const float* __restrict__ Wslice,
                                              unsigned lds_dst, unsigned voff) {
    asm volatile("global_load_async_to_lds_b128 %0, %1, %2"
                 :: "v"(lds_dst), "v"(voff), "s"(Wslice) : "memory");
    asm volatile("global_load_async_to_lds_b128 %0, %1, %2 offset:4096"
                 :: "v"(lds_dst), "v"(voff), "s"(Wslice) : "memory");
}

// ---------------- Kernel 4: 27-offset gather-GEMM via WMMA -----------------
// Block = 256 threads = 8 waves; each wave owns a 16-row x 64-col output tile
// (4 x v8f accumulators). W[k] slices are DMA'd into double-buffered LDS with
// GLOBAL_LOAD_ASYNC_TO_LDS_B128 so staging overlaps the WMMA stream.
__global__ __launch_bounds__(256) void spconv_wmma(const float* __restrict__ xn,
                                                   const float* __restrict__ W,
                                                   const int* __restrict__ nbr,
                                                   float* __restrict__ out,
                                                   int N) {
    __shared__ float ldsW[2][C_IN * C_OUT];  // 2 x 8 KB, row-major [k][n]

    const int tid  = threadIdx.x;
    const int wave = tid >> 5;
    const int lane = tid & 31;
    const int h    = lane >> 4;   // lane half: K sub-split for A/B fragments
    const int j    = lane & 15;   // M (voxel row) for A, N (C_OUT col) for B/D
    const int tb   = blockIdx.x * 128 + wave * 16;  // tile row base
    const int myrow = tb + j;
    const bool rowok = myrow < N;
    const int rsafe = rowok ? myrow : 0;

    // LDS byte offsets of the two buffers (flat shared addr low 32 bits = LDS
    // offset per the gfx1250 aperture mapping), plus this thread's 16B chunk.
    const unsigned voff  = (unsigned)tid * 16u;
    const unsigned ldsb0 = (unsigned)(size_t)&ldsW[0][0] + voff;
    const unsigned ldsb1 = (unsigned)(size_t)&ldsW[1][0] + voff;

    v8f acc0 = {}, acc1 = {}, acc2 = {}, acc3 = {};

    // Prologue: DMA slice 0 into buffer 0.
    stage_w_async(W, ldsb0, voff);

    for (int w = 0; w < KOFF; ++w) {
        // Own wave's DMA for buffer (w&1) has landed; barrier makes all waves'
        // chunks visible and retires all readers of buffer ((w+1)&1).
        asm volatile("s_wait_asynccnt 0x0" ::: "memory");
        __syncthreads();

        if (w + 1 < KOFF)  // overlap next slice's DMA with this slice's WMMAs
            stage_w_async(W + (size_t)(w + 1) * C_IN * C_OUT,
                          ((w + 1) & 1) ? ldsb1 : ldsb0, voff);

        const float* buf = &ldsW[w & 1][0];

        const int idx = nbr[(size_t)w * N + rsafe];
        const bool valid = rowok && (idx >= 0);
        const float mask = valid ? 1.0f : 0.0f;            // branchless: EXEC
        const int gr = valid ? idx : 0;                     // stays all-ones
        const float* arow = xn + (size_t)gr * C_IN + 2 * h; // A: K = v + 2h

#pragma unroll
        for (int kk = 0; kk < C_IN; kk += 4) {
            // A fragment: lane j holds x[idx[j]][kk + 2h + {0,1}]
            v2f a = *(const v2f*)(arow + kk);
            a = a * mask;  // zero contribution for missing neighbors

            const float* brow = &buf[(kk + 2 * h) * C_OUT + j];
            v2f b;

            b[0] = brow[0];              b[1] = brow[C_OUT + 0];
            acc0 = __builtin_amdgcn_wmma_f32_16x16x4_f32(
                false, a, false, b, (short)0, acc0, false, false);

            b[0] = brow[16];             b[1] = brow[C_OUT + 16];
            acc1 = __builtin_amdgcn_wmma_f32_16x16x4_f32(
                false, a, false, b, (short)0, acc1, false, false);

            b[0] = brow[32];             b[1] = brow[C_OUT + 32];
            acc2 = __builtin_amdgcn_wmma_f32_16x16x4_f32(
                false, a, false, b, (short)0, acc2, false, false);

            b[0] = brow[48];             b[1] = brow[C_OUT + 48];
            acc3 = __builtin_amdgcn_wmma_f32_16x16x4_f32(
                false, a, false, b, (short)0, acc3, false, false);
        }
    }

    // D layout: VGPR jj -> row M = jj + 8h, col N = nt*16 + j
#pragma unroll
    for (int jj = 0; jj < 8; ++jj) {
        const int row = tb + jj + 8 * h;
        if (row < N) {
            float* o = out + (size_t)row * C_OUT + j;
            o[0]  = acc0[jj];
            o[16] = acc1[jj];
            o[32] = acc2[jj];
            o[48] = acc3[jj];
        }
    }
}

// ---------------------------------------------------------------------------
extern "C" void kernel_launch(void* const* d_in, const int* in_sizes, int n_in,
                              void* d_out, int out_size, void* d_ws, size_t ws_size,
                              hipStream_t stream) {
    (void)n_in; (void)out_size; (void)ws_size;

    const float* features = (const float*)d_in[0];  // [N,32]
    const float* gamma    = (const float*)d_in[1];  // [32]
    const float* beta     = (const float*)d_in[2];  // [32]
    const float* W        = (const float*)d_in[3];  // [27,32,64]
    const int*   nbr      = (const int*)d_in[4];    // [27,N]
    float* out = (float*)d_out;                     // [N,64]

    const int N = in_sizes[0] / C_IN;

    // workspace layout (floats): sb[64] | partials[RB*64] | xn[N*32]
    float* ws       = (float*)d_ws;
    float* sb       = ws;
    float* partials = ws + 64;
    float* xn       = ws + 64 + RB * 64;   // byte offset 131328, 16B aligned

    bn_partials<<<RB, 256, 0, stream>>>(features, partials, N);
    bn_finalize<<<1, 32, 0, stream>>>(partials, gamma, beta, sb, N);

    const int nb_apply = (int)(((size_t)N * (C_IN / 4) + 255) / 256);
    bn_apply<<<nb_apply, 256, 0, stream>>>(features, sb, xn, N);

    const int nb_conv = (N + 127) / 128;
    spconv_wmma<<<nb_conv, 256, 0, stream>>>(xn, W, nbr, out, N);
}